// SAGEConv_66408784331089
// MI455X (gfx1250) — compile-verified
//
#include <hip/hip_runtime.h>
#include <math.h>

// ---------------------------------------------------------------------------
// MI455X (gfx1250, wave32) implementation of the SAGEConv-like reference.
// All 128x128 GEMMs (node-level and the dominant per-edge GEMMs) run on the
// fp32 WMMA pipe: V_WMMA_F32_16X16X4_F32, D = A(16x4) x B(4x16) + C(16x16).
// Weights are pre-normalized (weight-norm) and pre-packed into the exact
// B-fragment lane layout so each WMMA step is one coalesced v2f load.
// Edge kernel uses 8-way M-blocking (one B fragment feeds 8 WMMAs) and
// CDNA5 async global->LDS copies (GLOBAL_LOAD_ASYNC_TO_LDS_B128, ASYNCcnt)
// for the per-edge feature gathers.
// ---------------------------------------------------------------------------

typedef __attribute__((ext_vector_type(2))) float v2f;
typedef __attribute__((ext_vector_type(8))) float v8f;

#define LSTR 132   // LDS row stride (floats): 528B = 16B-aligned rows (async B128)
                   // and conflict-free: bank = (4m + k) mod 64 distinct per half-wave
#define ET   8     // edge tiles (of 16 edges) per block in edge_kernel

// CDNA5 async global->LDS copy: per-lane 16B DMA, tracked by ASYNCcnt.
// vdst VGPR holds the LDS byte address, vaddr the 64-bit global address.
__device__ __forceinline__ void async_copy_b128(const void* gptr, void* lds_ptr) {
    unsigned lds_addr = (unsigned)(size_t)lds_ptr;   // low 32 bits = LDS offset
    asm volatile("global_load_async_to_lds_b128 %0, %1, off"
                 :: "v"(lds_addr), "v"(gptr) : "memory");
}
__device__ __forceinline__ void async_wait_all() {
    asm volatile("s_wait_asynccnt 0" ::: "memory");
}

__device__ __forceinline__ float gelu_exact(float x) {
    // exact erf-based GELU (matches jax.nn.gelu(approximate=False))
    return 0.5f * x * (1.0f + erff(x * 0.70710678118654752440f));
}

__device__ __forceinline__ void atomicMaxFloat(float* addr, float val) {
    // IEEE-order trick: works for all sign mixes, init value is -inf.
    if (val >= 0.0f) atomicMax((int*)addr, __float_as_int(val));
    else             atomicMin((unsigned int*)addr, __float_as_uint(val));
}

// One 16(node) x 16(out) tile GEMM over K=128 in 32 steps of K=4.
// X  : LDS tile, 16 rows x 128 cols, row stride LSTR.
// Bp : packed B fragments for this layer (8192 v2f = 8 otiles x 32 ks x 32 lanes).
__device__ __forceinline__ v8f wmma_k128(const float* X, const float* Bp,
                                         int otile, int lane, v8f c) {
    const v2f* bf = (const v2f*)Bp + otile * 1024 + lane;
    const int mm   = lane & 15;
    const int koff = (lane >> 4) << 1;     // lanes 16..31 hold K+2,K+3
#pragma unroll
    for (int ks = 0; ks < 32; ++ks) {
        const int kb = (ks << 2) + koff;
        v2f a;
        a.x = X[mm * LSTR + kb];
        a.y = X[mm * LSTR + kb + 1];
        c = __builtin_amdgcn_wmma_f32_16x16x4_f32(false, a, false, bf[ks * 32],
                                                  (short)0, c, false, false);
    }
    return c;
}

// ---------------------------------------------------------------------------
// Weight preparation: weight-norm + pack into WMMA B-fragment layout.
// Layer order in Bpack: 0 asub, 1 amul, 2 asrc, 3 adst, 4 pool, 5 pool2,
//                       6 self, 7 neigh, 8 neigh2, 9 mlp0, 10 mlp1
// ---------------------------------------------------------------------------
struct WNLayers {
    const float* v[11];
    const float* g[11];
};

__global__ void prep_weights(WNLayers L, float* __restrict__ Bpack) {
    const int layer = blockIdx.x;
    const float* v = L.v[layer];
    const float* g = L.g[layer];
    __shared__ float scale[128];
    const int t = threadIdx.x;                 // 128 threads, one output row each
    float ss = 0.0f;
    const float* row = v + t * 128;
    for (int i = 0; i < 128; ++i) { float x = row[i]; ss += x * x; }
    scale[t] = g[t] / sqrtf(ss);
    __syncthreads();
    float* pack = Bpack + (size_t)layer * 16384;
    for (int f = t; f < 8192; f += 128) {      // fragment index
        const int lane = f & 31;
        const int ks   = (f >> 5) & 31;
        const int ot   = f >> 10;
        const int o    = ot * 16 + (lane & 15);
        const int kb   = (ks << 2) + ((lane >> 4) << 1);
        const float s  = scale[o];
        pack[2 * f + 0] = s * v[o * 128 + kb];
        pack[2 * f + 1] = s * v[o * 128 + kb + 1];
    }
}

__global__ void prep_aout(const float* __restrict__ v, const float* __restrict__ g,
                          float* __restrict__ waout) {
    __shared__ float red[128];
    const int t = threadIdx.x;
    const float x = v[t];
    red[t] = x * x;
    __syncthreads();
    for (int s = 64; s > 0; s >>= 1) { if (t < s) red[t] += red[t + s]; __syncthreads(); }
    waout[t] = g[0] * x / sqrtf(red[0]);
}

__global__ void init_buffers(float* __restrict__ neigh, float* __restrict__ neigh2,
                             float* __restrict__ cnt, int n128, int n) {
    const int i = blockIdx.x * blockDim.x + threadIdx.x;
    if (i < n128) { neigh[i] = -__builtin_inff(); neigh2[i] = 0.0f; }
    if (i < n) cnt[i] = 0.0f;
}

// ---------------------------------------------------------------------------
// Node pre-pass: xs = wn(feat,asrc), xd = wn(feat,adst),
//                h = gelu(wn(feat,pool)), h2 = gelu(wn(feat,pool2))
// 16-node tile per block, 4 waves, 4 layers x 8 otiles = 32 WMMA tile-tasks.
// ---------------------------------------------------------------------------
__global__ void node_pre(const float* __restrict__ feat, const float* __restrict__ Bpack,
                         const float* __restrict__ basrc, const float* __restrict__ badst,
                         const float* __restrict__ bpool, const float* __restrict__ bpool2,
                         float* __restrict__ xs, float* __restrict__ xd,
                         float* __restrict__ h, float* __restrict__ h2) {
    __shared__ float F[16 * LSTR];
    const int base = blockIdx.x * 16;
    const int tid  = threadIdx.x;
    {   // async-DMA the 16x128 feat tile into LDS (8 threads per row, b128 each)
        const int m = tid >> 3, l8 = tid & 7;
        const float* grow = feat + (size_t)(base + m) * 128;
#pragma unroll
        for (int c4 = 0; c4 < 4; ++c4) {
            const int col = l8 * 16 + c4 * 4;
            async_copy_b128(grow + col, &F[m * LSTR + col]);
        }
    }
    async_wait_all();
    __syncthreads();
    const int wave = tid >> 5, lane = tid & 31;
    for (int task = wave; task < 32; task += 4) {
        const int lsel  = task >> 3;     // 0:asrc 1:adst 2:pool 3:pool2
        const int otile = task & 7;
        const int layer = 2 + lsel;      // Bpack index
        v8f c = {};
        c = wmma_k128(F, Bpack + (size_t)layer * 16384, otile, lane, c);
        const float* bias = (lsel == 0) ? basrc : (lsel == 1) ? badst
                          : (lsel == 2) ? bpool : bpool2;
        float* outp = (lsel == 0) ? xs : (lsel == 1) ? xd : (lsel == 2) ? h : h2;
        const bool dog = (lsel >= 2);
        const int n = lane & 15, mh = (lane >> 4) << 3;
#pragma unroll
        for (int r = 0; r < 8; ++r) {
            const int m = r + mh;
            const int o = otile * 16 + n;
            float val = c[r] + bias[o];
            if (dog) val = gelu_exact(val);
            outp[(size_t)(base + m) * 128 + o] = val;
        }
    }
}

// ---------------------------------------------------------------------------
// Edge pass: per 128-edge block (8 tiles of 16 edges; one otile per wave)
//   e = gelu( (fs-fd)@Wsub^T + (fs*fd)@Wmul^T + xs[src] + xd[dst] + biases )
//   esc = leaky_relu(e . waout + aout_b, 0.2)
//   atomicMax(neigh[dst], esc*h[src]); atomicAdd(neigh2[dst], esc*h2[src]); cnt[dst]++
// Gathers go global->LDS via async B128 copies; the K-loop loads each packed
// B fragment once and feeds 8 WMMAs (8 live accumulators).
// ---------------------------------------------------------------------------
__global__ __launch_bounds__(256) void
edge_kernel(const float* __restrict__ feat,
            const int* __restrict__ src, const int* __restrict__ dst,
            const float* __restrict__ Bpack, const float* __restrict__ waout,
            const float* __restrict__ bsub, const float* __restrict__ bmul,
            const float* __restrict__ bsrc, const float* __restrict__ bdst,
            const float* __restrict__ aout_b,
            const float* __restrict__ xs, const float* __restrict__ xd,
            const float* __restrict__ h, const float* __restrict__ h2,
            float* __restrict__ neigh, float* __restrict__ neigh2,
            float* __restrict__ cnt) {
    __shared__ float U [ET * 16 * LSTR];   // async dst: fs rows; then fs - fd
    __shared__ float Mt[ET * 16 * LSTR];   // async dst: fd rows; then fs * fd
    __shared__ int   ssrc[ET * 16], sdst[ET * 16];
    __shared__ float earr[ET * 16], sesc[ET * 16];
    const int tid   = threadIdx.x;
    const int ebase = blockIdx.x * (ET * 16);      // 128 edges per block
    if (tid < ET * 16) {
        ssrc[tid] = src[ebase + tid];
        sdst[tid] = dst[ebase + tid];
        earr[tid] = 0.0f;
    }
    __syncthreads();
    // async-DMA gathers: 128 edges x 32 b128 chunks x 2 rows (src,dst)
    for (int ch = tid; ch < ET * 16 * 32; ch += 256) {
        const int m   = ch >> 5;           // edge within block
        const int col = (ch & 31) << 2;    // feature column (multiple of 4)
        async_copy_b128(feat + (size_t)ssrc[m] * 128 + col, &U [m * LSTR + col]);
        async_copy_b128(feat + (size_t)sdst[m] * 128 + col, &Mt[m * LSTR + col]);
    }
    async_wait_all();
    __syncthreads();
    // in-place transform: U <- fs - fd, Mt <- fs * fd
    for (int ch = tid; ch < ET * 16 * 32; ch += 256) {
        const int m   = ch >> 5;
        const int col = (ch & 31) << 2;
        float* u = &U [m * LSTR + col];
        float* w = &Mt[m * LSTR + col];
#pragma unroll
        for (int j = 0; j < 4; ++j) {
            const float a = u[j], b = w[j];
            u[j] = a - b;
            w[j] = a * b;
        }
    }
    __syncthreads();
    const int wave = tid >> 5, lane = tid & 31;
    const int otile = wave;                         // 8 waves <-> 8 output tiles
    const int mm   = lane & 15;
    const int koff = (lane >> 4) << 1;
    v8f c[ET];
    {
        v8f z = {};
#pragma unroll
        for (int t = 0; t < ET; ++t) c[t] = z;
    }
#pragma unroll
    for (int layer = 0; layer < 2; ++layer) {       // 0: asub on U, 1: amul on Mt
        const v2f* bf = (const v2f*)(Bpack + (size_t)layer * 16384) + otile * 1024 + lane;
        const float* X = (layer == 0) ? U : Mt;
#pragma unroll 4
        for (int ks = 0; ks < 32; ++ks) {
            const v2f b = bf[ks * 32];              // one B fragment, 8x reuse
            const int kb = (ks << 2) + koff;
#pragma unroll
            for (int t = 0; t < ET; ++t) {
                const float* Xt = X + t * 16 * LSTR;
                v2f a;
                a.x = Xt[mm * LSTR + kb];
                a.y = Xt[mm * LSTR + kb + 1];
                c[t] = __builtin_amdgcn_wmma_f32_16x16x4_f32(false, a, false, b,
                                                             (short)0, c[t], false, false);
            }
        }
    }
    // epilogue: add per-node terms + biases, gelu, dot with waout, reduce to earr
    const int n = lane & 15, mh = (lane >> 4) << 3;
    const int o = otile * 16 + n;
    const float wo   = waout[o];
    const float bsum = bsub[o] + bmul[o] + bsrc[o] + bdst[o];
#pragma unroll
    for (int t = 0; t < ET; ++t) {
#pragma unroll
        for (int r = 0; r < 8; ++r) {
            const int m = t * 16 + r + mh;
            float val = c[t][r] + xs[(size_t)ssrc[m] * 128 + o]
                                + xd[(size_t)sdst[m] * 128 + o] + bsum;
            float contrib = gelu_exact(val) * wo;
            for (int off = 8; off >= 1; off >>= 1)
                contrib += __shfl_xor(contrib, off, 16);
            if (n == 0) atomicAdd(&earr[m], contrib);
        }
    }
    __syncthreads();
    if (tid < ET * 16) {
        float e = earr[tid] + aout_b[0];
        sesc[tid] = (e > 0.0f) ? e : 0.2f * e;      // leaky_relu 0.2
        atomicAdd(&cnt[sdst[tid]], 1.0f);           // in-degree
    }
    __syncthreads();
    // scatter: half the block does segment-max, half does segment-sum
    const int colx = tid & 127;
    const int half = tid >> 7;
    for (int m = 0; m < ET * 16; ++m) {
        const float es = sesc[m];
        const size_t so  = (size_t)ssrc[m] * 128 + colx;
        const size_t dof = (size_t)sdst[m] * 128 + colx;
        if (half == 0) atomicMaxFloat(&neigh[dof], es * h[so]);
        else           atomicAdd(&neigh2[dof], es * h2[so]);
    }
}

// ---------------------------------------------------------------------------
// Final node pass: rst = wn(feat,self) + wn(neigh',neigh) + wn(neigh2',neigh2)
//                  then 2x residual MLP: rst += wn(gelu(rst), mlp_i)
// ---------------------------------------------------------------------------
__global__ void node_final(const float* __restrict__ feat, const float* __restrict__ neigh,
                           const float* __restrict__ neigh2, const float* __restrict__ cnt,
                           const float* __restrict__ Bpack,
                           const float* __restrict__ bself, const float* __restrict__ bneigh,
                           const float* __restrict__ bneigh2, const float* __restrict__ mlp_b,
                           float* __restrict__ out) {
    __shared__ float F [16 * LSTR];
    __shared__ float N1[16 * LSTR];
    __shared__ float N2[16 * LSTR];
    __shared__ float R [16 * LSTR];
    __shared__ float G [16 * LSTR];
    const int base = blockIdx.x * 16;
    const int tid  = threadIdx.x;
    {
        const int m = tid >> 3, l8 = tid & 7;
        const size_t row = (size_t)(base + m) * 128;
        // feat tile is a raw copy: async DMA it while we do the gated loads below
#pragma unroll
        for (int c4 = 0; c4 < 4; ++c4) {
            const int col = l8 * 16 + c4 * 4;
            async_copy_b128(feat + row + col, &F[m * LSTR + col]);
        }
        const float cv  = cnt[base + m];
        const float inv = 1.0f / fmaxf(cv, 1.0f);
        const bool  has = cv > 0.0f;
#pragma unroll
        for (int c4 = 0; c4 < 4; ++c4) {
            const int col = l8 * 16 + c4 * 4;
#pragma unroll
            for (int j = 0; j < 4; ++j) {
                const float nv = neigh[row + col + j];
                N1[m * LSTR + col + j] = has ? nv : 0.0f;     // -inf -> 0 for isolated
                N2[m * LSTR + col + j] = neigh2[row + col + j] * inv;
            }
        }
    }
    async_wait_all();
    __syncthreads();
    const int wave = tid >> 5, lane = tid & 31;
    for (int otile = wave; otile < 8; otile += 4) {
        v8f c = {};
        c = wmma_k128(F,  Bpack + 6  * 16384, otile, lane, c);   // self
        c = wmma_k128(N1, Bpack + 7  * 16384, otile, lane, c);   // neigh
        c = wmma_k128(N2, Bpack + 8  * 16384, otile, lane, c);   // neigh2
        const int n = lane & 15, mh = (lane >> 4) << 3;
        const int o = otile * 16 + n;
        const float bb = bself[o] + bneigh[o] + bneigh2[o];
#pragma unroll
        for (int r = 0; r < 8; ++r) R[(r + mh) * LSTR + o] = c[r] + bb;
    }
    __syncthreads();
    for (int it = 0; it < 2; ++it) {                              // residual MLP
        for (int q = tid; q < 2048; q += 128) {
            const int m = q >> 7, o = q & 127;
            G[m * LSTR + o] = gelu_exact(R[m * LSTR + o]);
        }
        __syncthreads();
        for (int otile = wave; otile < 8; otile += 4) {
            v8f c = {};
            c = wmma_k128(G, Bpack + (size_t)(9 + it) * 16384, otile, lane, c);
            const int n = lane & 15, mh = (lane >> 4) << 3;
            const int o = otile * 16 + n;
            const float bb = mlp_b[it * 128 + o];
#pragma unroll
            for (int r = 0; r < 8; ++r) R[(r + mh) * LSTR + o] += c[r] + bb;
        }
        __syncthreads();
    }
    for (int q = tid; q < 2048; q += 128) {
        const int m = q >> 7, o = q & 127;
        out[(size_t)(base + m) * 128 + o] = R[m * LSTR + o];
    }
}

// ---------------------------------------------------------------------------
extern "C" void kernel_launch(void* const* d_in, const int* in_sizes, int n_in,
                              void* d_out, int out_size, void* d_ws, size_t ws_size,
                              hipStream_t stream) {
    const float* feat    = (const float*)d_in[0];
    const float* asrc_v  = (const float*)d_in[1];
    const float* asrc_g  = (const float*)d_in[2];
    const float* asrc_b  = (const float*)d_in[3];
    const float* adst_v  = (const float*)d_in[4];
    const float* adst_g  = (const float*)d_in[5];
    const float* adst_b  = (const float*)d_in[6];
    const float* asub_v  = (const float*)d_in[7];
    const float* asub_g  = (const float*)d_in[8];
    const float* asub_b  = (const float*)d_in[9];
    const float* amul_v  = (const float*)d_in[10];
    const float* amul_g  = (const float*)d_in[11];
    const float* amul_b  = (const float*)d_in[12];
    const float* aout_v  = (const float*)d_in[13];
    const float* aout_g  = (const float*)d_in[14];
    const float* aout_b  = (const float*)d_in[15];
    const float* pool_v  = (const float*)d_in[16];
    const float* pool_g  = (const float*)d_in[17];
    const float* pool_b  = (const float*)d_in[18];
    const float* pool2_v = (const float*)d_in[19];
    const float* pool2_g = (const float*)d_in[20];
    const float* pool2_b = (const float*)d_in[21];
    const float* self_v  = (const float*)d_in[22];
    const float* self_g  = (const float*)d_in[23];
    const float* self_b  = (const float*)d_in[24];
    const float* neigh_v = (const float*)d_in[25];
    const float* neigh_g = (const float*)d_in[26];
    const float* neigh_b = (const float*)d_in[27];
    const float* neigh2_v= (const float*)d_in[28];
    const float* neigh2_g= (const float*)d_in[29];
    const float* neigh2_b= (const float*)d_in[30];
    const float* mlp_v   = (const float*)d_in[31];
    const float* mlp_g   = (const float*)d_in[32];
    const float* mlp_b   = (const float*)d_in[33];
    const int*   src     = (const int*)d_in[34];
    const int*   dst     = (const int*)d_in[35];
    float* out = (float*)d_out;

    const int N = in_sizes[0] / 128;   // 50000 (divisible by 16)
    const int E = in_sizes[34];        // 800000 (divisible by 128)

    // workspace layout (floats)
    float* ws    = (float*)d_ws;
    float* Bpack = ws;                                   // 11 * 16384
    float* waout = Bpack + (size_t)11 * 16384;           // 128
    float* xs    = waout + 128;                          // N*128 each below
    float* xd    = xs  + (size_t)N * 128;
    float* hbuf  = xd  + (size_t)N * 128;
    float* h2buf = hbuf+ (size_t)N * 128;
    float* nb    = h2buf + (size_t)N * 128;              // neigh (segment_max)
    float* n2b   = nb  + (size_t)N * 128;                // neigh2 (segment_sum)
    float* cntb  = n2b + (size_t)N * 128;                // in-degree

    WNLayers L;
    L.v[0] = asub_v;  L.g[0] = asub_g;
    L.v[1] = amul_v;  L.g[1] = amul_g;
    L.v[2] = asrc_v;  L.g[2] = asrc_g;
    L.v[3] = adst_v;  L.g[3] = adst_g;
    L.v[4] = pool_v;  L.g[4] = pool_g;
    L.v[5] = pool2_v; L.g[5] = pool2_g;
    L.v[6] = self_v;  L.g[6] = self_g;
    L.v[7] = neigh_v; L.g[7] = neigh_g;
    L.v[8] = neigh2_v;L.g[8] = neigh2_g;
    L.v[9] = mlp_v;           L.g[9]  = mlp_g;
    L.v[10]= mlp_v + 16384;   L.g[10] = mlp_g + 128;

    prep_weights<<<11, 128, 0, stream>>>(L, Bpack);
    prep_aout<<<1, 128, 0, stream>>>(aout_v, aout_g, waout);
    init_buffers<<<(N * 128 + 255) / 256, 256, 0, stream>>>(nb, n2b, cntb, N * 128, N);
    node_pre<<<N / 16, 128, 0, stream>>>(feat, Bpack, asrc_b, adst_b, pool_b, pool2_b,
                                         xs, xd, hbuf, h2buf);
    edge_kernel<<<E / (ET * 16), 256, 0, stream>>>(feat, src, dst, Bpack, waout,
                                                   asub_b, amul_b, asrc_b, adst_b, aout_b,
                                                   xs, xd, hbuf, h2buf, nb, n2b, cntb);
    node_final<<<N / 16, 128, 0, stream>>>(feat, nb, n2b, cntb, Bpack,
                                           self_b, neigh_b, neigh2_b, mlp_b, out);
    (void)n_in; (void)out_size; (void)ws_size;
}